// MessagePassingModel_7318624272528
// MI455X (gfx1250) — compile-verified
//
// CDNA5 (gfx1250 / MI455X) equivariant message-passing energy+forces model
// (forward + hand-derived backward wrt positions).
//
// Round-2 optimizations:
//  * All GEMMs use v_wmma_f32_16x16x4_f32; weight (B) operands are staged once
//    per block into LDS in a packed [k/4][n][4] layout so each WMMA B-pair is
//    one conflict-free ds_load_b64 (was 2x global_load_b32 per wave).
//  * A operands loaded as float2 (global_load_b64).
//  * Message-pass / elementwise kernels vectorized to float4 (global b128),
//    one thread per (edge, 4 features) — these are the HBM-roofline kernels.
//  * dsh per-edge reductions via wave32 __shfl_xor over 16-lane groups.

#include <hip/hip_runtime.h>

#define N_ATOMS  8000
#define N_EDGES  128000
#define FEAT     64
#define NBASIS   32
#define NLM      9
#define CUTOFF_R 3.0f
#define BATCH    8

#define NF   ((size_t)N_ATOMS * FEAT)        // one l-plane
#define LMNF ((size_t)NLM * N_ATOMS * FEAT)  // full feature tensor

typedef float v2f __attribute__((ext_vector_type(2)));
typedef float v4f __attribute__((ext_vector_type(4)));
typedef float v8f __attribute__((ext_vector_type(8)));

// comb(31, k)
__device__ __constant__ float BINOM31[32] = {
    1.f, 31.f, 465.f, 4495.f, 31465.f, 169911.f, 736281.f, 2629575.f,
    7888725.f, 20160075.f, 44352165.f, 84672315.f, 141120525.f, 206253075.f,
    265182525.f, 300540195.f, 300540195.f, 265182525.f, 206253075.f,
    141120525.f, 84672315.f, 44352165.f, 20160075.f, 7888725.f, 2629575.f,
    736281.f, 169911.f, 31465.f, 4495.f, 465.f, 31.f, 1.f};

// ---------------------------------------------------------------------------
// Packed-LDS WMMA helpers.
// Logical B is [K][N]; packed LDS layout: Bs[((k>>2)*N + n)*4 + (k&3)] so the
// (k0+kh, k0+kh+1) pair a lane needs is one aligned 8-byte ds load.
// A layout (ISA 7.12.2, 32-bit A 16x4): lane<16 -> v0/v1 = A[m][k0+0/1];
// lane>=16 -> A[m][k0+2/3].  C/D: vgpr i = row i (lanes 0-15) / 8+i (16-31).
// ---------------------------------------------------------------------------
template <int K, int N, bool TRANSB>
__device__ __forceinline__ void stage_B(const float* __restrict__ B, int ldb, float* Bs)
{
    const int Cc = TRANSB ? K : N;  // source is row-major [R][Cc]
    for (int i = threadIdx.x; i < K * N; i += 256) {
        int r = i / Cc, c = i % Cc;
        float v = B[(size_t)r * ldb + c];
        int k = TRANSB ? c : r;
        int n = TRANSB ? r : c;
        Bs[((k >> 2) * N + n) * 4 + (k & 3)] = v;
    }
}

template <int K, int N>
__device__ __forceinline__ v8f wave_gemm16(const float* __restrict__ Arow, int lda,
                                           const float* Bs, int n0, int lane, v8f acc)
{
    const int m  = lane & 15;
    const int kh = (lane >> 4) << 1;  // 0 or 2
    const float* ap = Arow + (size_t)m * lda + kh;
    const float* bp = Bs + (size_t)(n0 + m) * 4 + kh;
#pragma unroll
    for (int k0 = 0; k0 < K; k0 += 4) {
        v2f a = *(const v2f*)(ap + k0);
        v2f b = *(const v2f*)(bp + (k0 >> 2) * (N * 4));
        acc = __builtin_amdgcn_wmma_f32_16x16x4_f32(false, a, false, b,
                                                    (short)0, acc, false, false);
    }
    return acc;
}

// C[MxN] = A[MxK] @ B (optionally B transposed, optionally accumulate into C).
// 8 waves / block, one 16-row tile per wave. M must be a multiple of 16.
template <int K, int N, bool TRANSB, bool ACCUM>
__global__ __launch_bounds__(256) void gemm_plain_kernel(const float* __restrict__ A,
                                                         const float* __restrict__ B, int ldb,
                                                         float* __restrict__ C, int M)
{
    __shared__ float Bs[K * N];
    stage_B<K, N, TRANSB>(B, ldb, Bs);
    __syncthreads();
    const int lane    = threadIdx.x & 31;
    const int tile    = blockIdx.x * 8 + (threadIdx.x >> 5);
    const int rowbase = tile * 16;
    if (rowbase < M) {  // wave-uniform (EXEC all-1 for WMMA)
        const float* Arow = A + (size_t)rowbase * K;
        float* Crow       = C + (size_t)rowbase * N;
        const int m  = lane & 15;
        const int cm = (lane < 16) ? 0 : 8;
#pragma unroll
        for (int n0 = 0; n0 < N; n0 += 16) {
            v8f acc = {};
            if (ACCUM) {
#pragma unroll
                for (int i = 0; i < 8; ++i) acc[i] = Crow[(size_t)(cm + i) * N + n0 + m];
            }
            acc = wave_gemm16<K, N>(Arow, K, Bs, n0, lane, acc);
#pragma unroll
            for (int i = 0; i < 8; ++i) Crow[(size_t)(cm + i) * N + n0 + m] = acc[i];
        }
    }
}

// dense_deg: rows are [l][node] (l-major). Weight selected by degree of l,
// bias added on l==0 rows, optional residual add. All 3 degree matrices are
// staged packed into LDS (48 KB).
template <bool TRANSB>
__global__ __launch_bounds__(256) void gemm_dense_kernel(const float* __restrict__ X,
                                                         const float* __restrict__ W3,
                                                         const float* __restrict__ bias,
                                                         const float* __restrict__ resid,
                                                         float* __restrict__ Y)
{
    __shared__ float Bs[3 * FEAT * FEAT];
    for (int i = threadIdx.x; i < 3 * FEAT * FEAT; i += 256) {
        int deg = i >> 12, j = i & 4095, r = j >> 6, c = j & 63;
        int k = TRANSB ? c : r;
        int n = TRANSB ? r : c;
        Bs[deg * 4096 + ((k >> 2) * FEAT + n) * 4 + (k & 3)] = W3[i];
    }
    __syncthreads();
    const int lane    = threadIdx.x & 31;
    const int tile    = blockIdx.x * 8 + (threadIdx.x >> 5);
    const int rowbase = tile * 16;
    if (rowbase < NLM * N_ATOMS) {  // wave-uniform
        const int l   = rowbase / N_ATOMS;  // N_ATOMS % 16 == 0 -> tile-uniform
        const int deg = (l == 0) ? 0 : ((l < 4) ? 1 : 2);
        const float* Bsd  = Bs + deg * 4096;
        const float* Arow = X + (size_t)rowbase * FEAT;
        float* Crow       = Y + (size_t)rowbase * FEAT;
        const float* Rrow = resid ? resid + (size_t)rowbase * FEAT : nullptr;
        const bool addb   = (bias != nullptr) && (l == 0);
        const int m  = lane & 15;
        const int cm = (lane < 16) ? 0 : 8;
#pragma unroll
        for (int n0 = 0; n0 < FEAT; n0 += 16) {
            v8f acc = {};
            acc = wave_gemm16<FEAT, FEAT>(Arow, FEAT, Bsd, n0, lane, acc);
            const float bv = addb ? bias[n0 + m] : 0.0f;
#pragma unroll
            for (int i = 0; i < 8; ++i) {
                float v = acc[i] + bv;
                if (Rrow) v += Rrow[(size_t)(cm + i) * FEAT + n0 + m];
                Crow[(size_t)(cm + i) * FEAT + n0 + m] = v;
            }
        }
    }
}

// ---------------------------------------------------------------------------
// Edge geometry: d = pos[src]-pos[dst], r, u, spherical harmonics, radial basis
// ---------------------------------------------------------------------------
__global__ void edge_geom_kernel(const float* __restrict__ pos, const int* __restrict__ srcI,
                                 const int* __restrict__ dstI, float* __restrict__ sh,
                                 float* __restrict__ rad, float* __restrict__ uvec,
                                 float* __restrict__ rout)
{
    int e = blockIdx.x * blockDim.x + threadIdx.x;
    if (e >= N_EDGES) return;
    int s = srcI[e], d = dstI[e];
    float dx = pos[3 * s + 0] - pos[3 * d + 0];
    float dy = pos[3 * s + 1] - pos[3 * d + 1];
    float dz = pos[3 * s + 2] - pos[3 * d + 2];
    float r  = sqrtf(dx * dx + dy * dy + dz * dz + 1e-12f);
    float iv = 1.0f / r;
    float x = dx * iv, y = dy * iv, z = dz * iv;
    const float c0 = 0.28209479177387814f, c1 = 0.4886025119029199f;
    const float c2a = 1.0925484305920792f, c2b = 0.31539156525252005f, c2c = 0.5462742152960396f;
    float* she = sh + (size_t)e * NLM;
    she[0] = c0;          she[1] = c1 * y;      she[2] = c1 * z;
    she[3] = c1 * x;      she[4] = c2a * x * y; she[5] = c2a * y * z;
    she[6] = c2b * (3.f * z * z - 1.f);         she[7] = c2a * x * z;
    she[8] = c2c * (x * x - y * y);
    uvec[3 * e + 0] = x; uvec[3 * e + 1] = y; uvec[3 * e + 2] = z;
    rout[e] = r;
    float s2  = (r / CUTOFF_R) * (r / CUTOFF_R);
    float cut = (s2 < 1.f) ? expf(1.f - 1.f / (1.f - s2)) : 0.f;
    float t  = 1.f / (1.f + r);
    float lt = logf(t), lo = logf(1.f - t);
    float* re = rad + (size_t)e * NBASIS;
    for (int k = 0; k < NBASIS; ++k)
        re[k] = BINOM31[k] * expf((float)k * lt + (float)(NBASIS - 1 - k) * lo) * cut;
}

__global__ void init_x_kernel(const int* __restrict__ Z, const float* __restrict__ embed,
                              float* __restrict__ x)
{
    int idx = blockIdx.x * 256 + threadIdx.x;  // over NF/4
    if (idx >= (int)(NF / 4)) return;
    int n = idx >> 4, fb = (idx & 15) << 2;
    *(v4f*)(x + ((size_t)idx << 2)) = *(const v4f*)(embed + (size_t)Z[n] * FEAT + fb);
    v4f zero = {};
#pragma unroll
    for (int l = 1; l < NLM; ++l) *(v4f*)(x + (size_t)l * NF + ((size_t)idx << 2)) = zero;
}

// ---------------------------------------------------------------------------
// Message passing forward: y[dst] += m(e); one thread per (edge, 4 features)
// ---------------------------------------------------------------------------
__global__ __launch_bounds__(256) void msg_fwd_kernel(const float* __restrict__ x,
        const float* __restrict__ sh, const float* __restrict__ wA,
        const float* __restrict__ wB, const float* __restrict__ wC,
        const int* __restrict__ srcI, const int* __restrict__ dstI, float* __restrict__ y)
{
    int tid = blockIdx.x * 256 + threadIdx.x;
    int e = tid >> 4, fb = (tid & 15) << 2;
    if (e >= N_EDGES) return;
    int s = srcI[e], d = dstI[e];
    float shv[NLM];
#pragma unroll
    for (int l = 0; l < NLM; ++l) shv[l] = sh[(size_t)e * NLM + l];
    v4f wa = *(const v4f*)(wA + (size_t)e * FEAT + fb);
    v4f wb = *(const v4f*)(wB + (size_t)e * FEAT + fb);
    v4f wc = *(const v4f*)(wC + (size_t)e * FEAT + fb);
    v4f xs[NLM];
    v4f dotX = {};
#pragma unroll
    for (int l = 0; l < NLM; ++l) {
        xs[l] = *(const v4f*)(x + (size_t)l * NF + (size_t)s * FEAT + fb);
        dotX += xs[l] * shv[l];
    }
    v4f xb = xs[0] * wb;
#pragma unroll
    for (int l = 0; l < NLM; ++l) {
        v4f m = xs[l] * wa + shv[l] * xb;
        if (l == 0) m += dotX * wc;
        float* yp = y + (size_t)l * NF + (size_t)d * FEAT + fb;
        atomicAdd(yp + 0, m.x);
        atomicAdd(yp + 1, m.y);
        atomicAdd(yp + 2, m.z);
        atomicAdd(yp + 3, m.w);
    }
}

__global__ void silu_fwd_kernel(const float* __restrict__ t, float* __restrict__ g)
{
    int idx = blockIdx.x * 256 + threadIdx.x;  // over NF/4
    if (idx >= (int)(NF / 4)) return;
    size_t base = (size_t)idx << 2;
    v4f s = *(const v4f*)(t + base);
    v4f sig;
#pragma unroll
    for (int i = 0; i < 4; ++i) sig[i] = 1.f / (1.f + expf(-s[i]));
    *(v4f*)(g + base) = s * sig;
#pragma unroll
    for (int l = 1; l < NLM; ++l)
        *(v4f*)(g + (size_t)l * NF + base) = *(const v4f*)(t + (size_t)l * NF + base) * sig;
}

__global__ void energy_kernel(const float* __restrict__ xf, const float* __restrict__ wE,
                              const float* __restrict__ ebias, const int* __restrict__ Z,
                              const int* __restrict__ seg, float* __restrict__ energy)
{
    int n = blockIdx.x * 256 + threadIdx.x;
    if (n >= N_ATOMS) return;
    v4f acc = {};
    for (int f = 0; f < FEAT; f += 4)
        acc += *(const v4f*)(xf + (size_t)n * FEAT + f) * *(const v4f*)(wE + f);
    float a = ebias[Z[n]] + acc.x + acc.y + acc.z + acc.w;
    atomicAdd(&energy[seg[n]], a);
}

// ---------------------------------------------------------------------------
// Backward pieces
// ---------------------------------------------------------------------------
__global__ void grad_init_kernel(const float* __restrict__ wE, float* __restrict__ dX)
{
    int idx = blockIdx.x * 256 + threadIdx.x;  // over NF/4
    if (idx >= (int)(NF / 4)) return;
    size_t base = (size_t)idx << 2;
    v4f w = *(const v4f*)(wE + ((idx & 15) << 2));
    *(v4f*)(dX + base) = -w;  // dL/dx_final[:,0,:] ; L = -sum(energy)
    v4f zero = {};
#pragma unroll
    for (int l = 1; l < NLM; ++l) *(v4f*)(dX + (size_t)l * NF + base) = zero;
}

__global__ void silu_bwd_kernel(const float* __restrict__ dG, const float* __restrict__ t,
                                float* __restrict__ dT)
{
    int idx = blockIdx.x * 256 + threadIdx.x;  // over NF/4
    if (idx >= (int)(NF / 4)) return;
    size_t base = (size_t)idx << 2;
    v4f s = *(const v4f*)(t + base);
    v4f sig;
#pragma unroll
    for (int i = 0; i < 4; ++i) sig[i] = 1.f / (1.f + expf(-s[i]));
    v4f sigp = sig * (1.f - sig);
    v4f acc  = *(const v4f*)(dG + base) * (sig + s * sigp);
#pragma unroll
    for (int l = 1; l < NLM; ++l) {
        v4f dg = *(const v4f*)(dG + (size_t)l * NF + base);
        *(v4f*)(dT + (size_t)l * NF + base) = dg * sig;
        acc += dg * *(const v4f*)(t + (size_t)l * NF + base) * sigp;
    }
    *(v4f*)(dT + base) = acc;
}

__global__ void add_kernel(const float* __restrict__ a, const float* __restrict__ b,
                           float* __restrict__ c)
{
    size_t base = ((size_t)blockIdx.x * 256 + threadIdx.x) << 2;  // grid = LMNF/4/256
    *(v4f*)(c + base) = *(const v4f*)(a + base) + *(const v4f*)(b + base);
}

// Message backward: scatters dxs into dXn[src], reduces dsh per edge,
// overwrites wA/wB/wC in place with dwA/dwB/dwC (per-thread locations).
__global__ __launch_bounds__(256) void msg_bwd_kernel(const float* __restrict__ xin,
        const float* __restrict__ dY0, const float* __restrict__ sh,
        float* __restrict__ wA, float* __restrict__ wB, float* __restrict__ wC,
        const int* __restrict__ srcI, const int* __restrict__ dstI,
        float* __restrict__ dXn, float* __restrict__ dsh)
{
    int tid = blockIdx.x * 256 + threadIdx.x;
    int e = tid >> 4, fb = (tid & 15) << 2;
    if (e >= N_EDGES) return;
    int s = srcI[e], d = dstI[e];
    float shv[NLM];
#pragma unroll
    for (int l = 0; l < NLM; ++l) shv[l] = sh[(size_t)e * NLM + l];
    v4f wa = *(const v4f*)(wA + (size_t)e * FEAT + fb);
    v4f wb = *(const v4f*)(wB + (size_t)e * FEAT + fb);
    v4f wc = *(const v4f*)(wC + (size_t)e * FEAT + fb);
    v4f G[NLM], xs[NLM];
    v4f dotG = {}, dotX = {};
#pragma unroll
    for (int l = 0; l < NLM; ++l) {
        G[l]  = *(const v4f*)(dY0 + (size_t)l * NF + (size_t)d * FEAT + fb);
        xs[l] = *(const v4f*)(xin + (size_t)l * NF + (size_t)s * FEAT + fb);
        dotG += G[l] * shv[l];
        dotX += xs[l] * shv[l];
    }
#pragma unroll
    for (int l = 0; l < NLM; ++l) {
        v4f dx = G[l] * wa + G[0] * wc * shv[l];
        if (l == 0) dx += dotG * wb;
        float* dp = dXn + (size_t)l * NF + (size_t)s * FEAT + fb;
        atomicAdd(dp + 0, dx.x);
        atomicAdd(dp + 1, dx.y);
        atomicAdd(dp + 2, dx.z);
        atomicAdd(dp + 3, dx.w);
    }
    // dsh[l] = sum_f (G[l]*xs0*wB + G[0]*xs[l]*wC); 16-lane groups (2 edges/wave)
#pragma unroll
    for (int l = 0; l < NLM; ++l) {
        v4f cv = G[l] * xs[0] * wb + G[0] * xs[l] * wc;
        float c = cv.x + cv.y + cv.z + cv.w;
        c += __shfl_xor(c, 1, 32);
        c += __shfl_xor(c, 2, 32);
        c += __shfl_xor(c, 4, 32);
        c += __shfl_xor(c, 8, 32);
        if ((threadIdx.x & 15) == 0) atomicAdd(&dsh[(size_t)e * NLM + l], c);
    }
    v4f dwa = {};
#pragma unroll
    for (int l = 0; l < NLM; ++l) dwa += G[l] * xs[l];
    *(v4f*)(wA + (size_t)e * FEAT + fb) = dwa;
    *(v4f*)(wB + (size_t)e * FEAT + fb) = dotG * xs[0];
    *(v4f*)(wC + (size_t)e * FEAT + fb) = G[0] * dotX;
}

// Geometry backward: dsh -> du -> dd ; drad -> dr -> dd ; scatter into forces.
__global__ void geom_bwd_kernel(const float* __restrict__ uvec, const float* __restrict__ rr,
                                const float* __restrict__ dsh, const float* __restrict__ drad,
                                const int* __restrict__ srcI, const int* __restrict__ dstI,
                                float* __restrict__ forces)
{
    int e = blockIdx.x * blockDim.x + threadIdx.x;
    if (e >= N_EDGES) return;
    float x = uvec[3 * e + 0], y = uvec[3 * e + 1], z = uvec[3 * e + 2];
    float r = rr[e];
    const float c1 = 0.4886025119029199f, c2a = 1.0925484305920792f;
    const float c2b = 0.31539156525252005f, c2c = 0.5462742152960396f;
    const float* de = dsh + (size_t)e * NLM;
    float dux = c1 * de[3] + c2a * y * de[4] + c2a * z * de[7] + 2.f * c2c * x * de[8];
    float duy = c1 * de[1] + c2a * x * de[4] + c2a * z * de[5] - 2.f * c2c * y * de[8];
    float duz = c1 * de[2] + c2a * y * de[5] + 6.f * c2b * z * de[6] + c2a * x * de[7];
    float t   = 1.f / (1.f + r);
    float omt = 1.f - t;
    float s2  = (r / CUTOFF_R) * (r / CUTOFF_R);
    float cut = 0.f, dcut = 0.f;
    if (s2 < 1.f) {
        float om = 1.f - s2;
        cut  = expf(1.f - 1.f / om);
        dcut = -cut * (2.f * r / (CUTOFF_R * CUTOFF_R)) / (om * om);
    }
    float lt = logf(t), lo = logf(omt);
    float drtot = 0.f;
    const float* dre = drad + (size_t)e * NBASIS;
    for (int k = 0; k < NBASIS; ++k) {
        float P = BINOM31[k] * expf((float)k * lt + (float)(NBASIS - 1 - k) * lo);
        float q = -(float)k * t + (float)(NBASIS - 1 - k) * t * t / omt;  // dlnP/dr
        drtot += dre[k] * P * (cut * q + dcut);
    }
    float udot = x * dux + y * duy + z * duz;
    float invr = 1.f / r;
    float ddx = (dux - x * udot) * invr + x * drtot;
    float ddy = (duy - y * udot) * invr + y * drtot;
    float ddz = (duz - z * udot) * invr + z * drtot;
    int s = srcI[e], d = dstI[e];
    atomicAdd(&forces[3 * s + 0], ddx);
    atomicAdd(&forces[3 * s + 1], ddy);
    atomicAdd(&forces[3 * s + 2], ddz);
    atomicAdd(&forces[3 * d + 0], -ddx);
    atomicAdd(&forces[3 * d + 1], -ddy);
    atomicAdd(&forces[3 * d + 2], -ddz);
}

// ---------------------------------------------------------------------------
extern "C" void kernel_launch(void* const* d_in, const int* in_sizes, int n_in,
                              void* d_out, int out_size, void* d_ws, size_t ws_size,
                              hipStream_t stream)
{
    const int*   Z     = (const int*)d_in[0];
    const float* pos   = (const float*)d_in[1];
    const int*   dstI  = (const int*)d_in[2];
    const int*   srcI  = (const int*)d_in[3];
    const int*   seg   = (const int*)d_in[4];
    /* d_in[5] = batch_size scalar (compile-time BATCH=8) */
    const float* embed = (const float*)d_in[6];
    const float* ebias = (const float*)d_in[7];
    const float* WA    = (const float*)d_in[8];
    const float* WB    = (const float*)d_in[9];
    const float* WC    = (const float*)d_in[10];
    const float* D1    = (const float*)d_in[11];
    const float* b1    = (const float*)d_in[12];
    const float* D2    = (const float*)d_in[13];
    const float* b2    = (const float*)d_in[14];
    const float* wE    = (const float*)d_in[15];
    (void)in_sizes; (void)n_in; (void)out_size; (void)ws_size;

    float* Wp = (float*)d_ws;
    size_t off = 0;
    auto alloc = [&](size_t n) { float* p = Wp + off; off += n; return p; };
    float* sh   = alloc((size_t)N_EDGES * NLM);
    float* rad  = alloc((size_t)N_EDGES * NBASIS);
    float* uvec = alloc((size_t)N_EDGES * 3);
    float* rbuf = alloc((size_t)N_EDGES);
    float* x0 = alloc(LMNF);
    float* x1 = alloc(LMNF);
    float* x2 = alloc(LMNF);
    float* t0 = alloc(LMNF);
    float* t1 = alloc(LMNF);
    float* g  = alloc(LMNF);
    float* wAe = alloc((size_t)N_EDGES * FEAT);
    float* wBe = alloc((size_t)N_EDGES * FEAT);
    float* wCe = alloc((size_t)N_EDGES * FEAT);
    float* P = alloc(LMNF);   // rotating gradient buffers
    float* Q = alloc(LMNF);
    float* R = alloc(LMNF);
    float* dsh  = alloc((size_t)N_EDGES * NLM);
    float* drad = alloc((size_t)N_EDGES * NBASIS);  // total ~308 MB

    float* energy = (float*)d_out;
    float* forces = (float*)d_out + BATCH;

    hipMemsetAsync(d_out, 0, sizeof(float) * (BATCH + 3 * N_ATOMS), stream);

    edge_geom_kernel<<<(N_EDGES + 255) / 256, 256, 0, stream>>>(pos, srcI, dstI, sh, rad, uvec, rbuf);
    init_x_kernel<<<(int)(NF / 4 / 256), 256, 0, stream>>>(Z, embed, x0);

    const float* xin[2]  = {x0, x1};
    float*       xout[2] = {x1, x2};
    float*       tbuf[2] = {t0, t1};
    const int dense_blks = (NLM * N_ATOMS / 16 + 7) / 8;            // 563
    const int edge_blks  = N_EDGES / 16 / 8;                        // 1000
    const int msg_blks   = (int)((size_t)N_EDGES * 16 / 256);       // 8000
    const int nf4_blks   = (int)(NF / 4 / 256);                     // 500

    // ---------------- forward ----------------
    for (int it = 0; it < 2; ++it) {
        gemm_plain_kernel<NBASIS, FEAT, false, false><<<edge_blks, 256, 0, stream>>>(
            rad, WA + (size_t)it * NBASIS * FEAT, FEAT, wAe, N_EDGES);
        gemm_plain_kernel<NBASIS, FEAT, false, false><<<edge_blks, 256, 0, stream>>>(
            rad, WB + (size_t)it * NBASIS * FEAT, FEAT, wBe, N_EDGES);
        gemm_plain_kernel<NBASIS, FEAT, false, false><<<edge_blks, 256, 0, stream>>>(
            rad, WC + (size_t)it * NBASIS * FEAT, FEAT, wCe, N_EDGES);
        hipMemcpyAsync(P, xin[it], LMNF * sizeof(float), hipMemcpyDeviceToDevice, stream);
        msg_fwd_kernel<<<msg_blks, 256, 0, stream>>>(xin[it], sh, wAe, wBe, wCe, srcI, dstI, P);
        gemm_dense_kernel<false><<<dense_blks, 256, 0, stream>>>(
            P, D1 + (size_t)it * 3 * FEAT * FEAT, b1 + (size_t)it * FEAT, nullptr, tbuf[it]);
        silu_fwd_kernel<<<nf4_blks, 256, 0, stream>>>(tbuf[it], g);
        gemm_dense_kernel<false><<<dense_blks, 256, 0, stream>>>(
            g, D2 + (size_t)it * 3 * FEAT * FEAT, b2 + (size_t)it * FEAT, xin[it], xout[it]);
    }
    energy_kernel<<<(N_ATOMS + 255) / 256, 256, 0, stream>>>(x2, wE, ebias, Z, seg, energy);

    // ---------------- backward ----------------
    hipMemsetAsync(dsh, 0, (size_t)N_EDGES * NLM * sizeof(float), stream);
    hipMemsetAsync(drad, 0, (size_t)N_EDGES * NBASIS * sizeof(float), stream);
    grad_init_kernel<<<nf4_blks, 256, 0, stream>>>(wE, P);

    float* dX = P;
    float* bA = Q;
    float* bB = R;
    for (int it = 1; it >= 0; --it) {
        gemm_dense_kernel<true><<<dense_blks, 256, 0, stream>>>(
            dX, D2 + (size_t)it * 3 * FEAT * FEAT, nullptr, nullptr, bA);          // dG
        silu_bwd_kernel<<<nf4_blks, 256, 0, stream>>>(bA, tbuf[it], bB);            // dT
        gemm_dense_kernel<true><<<dense_blks, 256, 0, stream>>>(
            bB, D1 + (size_t)it * 3 * FEAT * FEAT, nullptr, nullptr, bA);          // dY0
        add_kernel<<<(int)(LMNF / 4 / 256), 256, 0, stream>>>(dX, bA, bB);         // dXn base
        gemm_plain_kernel<NBASIS, FEAT, false, false><<<edge_blks, 256, 0, stream>>>(
            rad, WA + (size_t)it * NBASIS * FEAT, FEAT, wAe, N_EDGES);
        gemm_plain_kernel<NBASIS, FEAT, false, false><<<edge_blks, 256, 0, stream>>>(
            rad, WB + (size_t)it * NBASIS * FEAT, FEAT, wBe, N_EDGES);
        gemm_plain_kernel<NBASIS, FEAT, false, false><<<edge_blks, 256, 0, stream>>>(
            rad, WC + (size_t)it * NBASIS * FEAT, FEAT, wCe, N_EDGES);
        msg_bwd_kernel<<<msg_blks, 256, 0, stream>>>(xin[it], bA, sh, wAe, wBe, wCe,
                                                     srcI, dstI, bB, dsh);
        gemm_plain_kernel<FEAT, NBASIS, true, true><<<edge_blks, 256, 0, stream>>>(
            wAe, WA + (size_t)it * NBASIS * FEAT, FEAT, drad, N_EDGES);
        gemm_plain_kernel<FEAT, NBASIS, true, true><<<edge_blks, 256, 0, stream>>>(
            wBe, WB + (size_t)it * NBASIS * FEAT, FEAT, drad, N_EDGES);
        gemm_plain_kernel<FEAT, NBASIS, true, true><<<edge_blks, 256, 0, stream>>>(
            wCe, WC + (size_t)it * NBASIS * FEAT, FEAT, drad, N_EDGES);
        float* nX = bB;  // rotate (P,Q,R) -> (R,P,Q)
        bB = bA;
        bA = dX;
        dX = nX;
    }
    geom_bwd_kernel<<<(N_EDGES + 255) / 256, 256, 0, stream>>>(uvec, rbuf, dsh, drad,
                                                               srcI, dstI, forces);
}